// Model_55499567399094
// MI455X (gfx1250) — compile-verified
//
#include <hip/hip_runtime.h>
#include <hip/hip_bf16.h>

typedef __attribute__((ext_vector_type(2))) float v2f;
typedef __attribute__((ext_vector_type(8))) float v8f;

#define BSZ 8
#define RNK 9
#define SLC 4
#define TOK 576
#define HID 1024
#define NROW (BSZ * RNK * SLC)          // 288
#define TT   (TOK / 16)                 // 36 t-tiles per n
#define NWAVE (NROW * TT)               // 10368 waves
#define WPB  8                          // waves per block (256 threads)

// ---------------------------------------------------------------------------
// Kernel 1: rewards[n,t] = dot(h[n,t,:], w) + vb   via V_WMMA_F32_16X16X4_F32
// One wave -> 16 rows (fixed n, 16 consecutive t). K loop: 1024 in steps of 8 (two K=4 WMMAs).
// A (16x4): lane l, m=l&15, kh=l>>4 holds h[row m][kb+2kh], h[row m][kb+2kh+1]
// B (4x16): all 16 columns carry the same weight chunk -> D[i][j]=dot for all j
// ---------------------------------------------------------------------------
__global__ __launch_bounds__(256) void k_rewards(
    const float* __restrict__ h, const float* __restrict__ w,
    const float* __restrict__ vb, float* __restrict__ rewards) {
  __shared__ __align__(16) float wl[HID];
  for (int i = threadIdx.x; i < HID; i += 256) wl[i] = w[i];
  __syncthreads();

  const int wave = threadIdx.x >> 5;
  const int lane = threadIdx.x & 31;
  const int gw   = blockIdx.x * WPB + wave;     // 0..10367
  const int n    = gw / TT;                     // 0..287
  const int t0   = (gw % TT) * 16;
  const int m    = lane & 15;                   // row within tile
  const int kh   = lane >> 4;                   // K-half select (0/1)

  const float* rp = h + ((size_t)n * TOK + t0 + m) * HID + 2 * kh;
  const float* wp = wl + 2 * kh;

  v8f acc0 = {};
  v8f acc1 = {};
  #pragma unroll 4
  for (int kb = 0; kb < HID; kb += 8) {
    v2f a0 = *(const v2f*)(rp + kb);
    v2f b0 = *(const v2f*)(wp + kb);
    acc0 = __builtin_amdgcn_wmma_f32_16x16x4_f32(
        false, a0, false, b0, (short)0, acc0, false, false);
    v2f a1 = *(const v2f*)(rp + kb + 4);
    v2f b1 = *(const v2f*)(wp + kb + 4);
    acc1 = __builtin_amdgcn_wmma_f32_16x16x4_f32(
        false, a1, false, b1, (short)0, acc1, false, false);
  }

  const float bias = vb[0];
  // D layout: VGPR v, lanes 0-15: M=v (N=lane); lanes 16-31: M=v+8.
  // Column N=0 of each half: lane 0 -> rows t0..t0+7, lane 16 -> rows t0+8..t0+15
  if (m == 0) {
    float* out = rewards + (size_t)n * TOK + t0 + kh * 8;
    #pragma unroll
    for (int v = 0; v < 8; ++v) out[v] = acc0[v] + acc1[v] + bias;
  }
}

// ---------------------------------------------------------------------------
// Kernel 2: ranked[br][t] = mean over 4 slices; end_scores[r*8+b] = mean_t
// One block per br = b*9+r (72 blocks, 576 threads).
// ---------------------------------------------------------------------------
__global__ __launch_bounds__(TOK) void k_ranked(
    const float* __restrict__ rewards, float* __restrict__ ranked,
    float* __restrict__ out) {
  const int br = blockIdx.x;          // b*9 + r
  const int t  = threadIdx.x;         // 0..575
  __shared__ float s[TOK];

  const float* p = rewards + (size_t)br * SLC * TOK + t;
  float v = 0.25f * (p[0] + p[TOK] + p[2 * TOK] + p[3 * TOK]);
  ranked[br * TOK + t] = v;
  s[t] = v;
  __syncthreads();
  for (int off = 512; off >= 1; off >>= 1) {
    if (t < off && t + off < TOK) s[t] += s[t + off];
    __syncthreads();
  }
  if (t == 0) {
    const int b = br / RNK, r = br % RNK;
    out[r * BSZ + b] = s[0] * (1.0f / TOK);
  }
}

// ---------------------------------------------------------------------------
// Kernel 3: loss = mean_b [ sum_{j<k} mean_t softplus(-(r_j - r_k)) / 36 ]
// One block, 576 threads; thread t handles all 8*36 pairs at its token.
// ---------------------------------------------------------------------------
__global__ __launch_bounds__(TOK) void k_loss(
    const float* __restrict__ ranked, float* __restrict__ out) {
  const int t = threadIdx.x;
  __shared__ float s[TOK];
  float acc = 0.0f;
  for (int b = 0; b < BSZ; ++b) {
    const float* rb = ranked + (size_t)b * RNK * TOK + t;
    float r[RNK];
    #pragma unroll
    for (int j = 0; j < RNK; ++j) r[j] = rb[j * TOK];
    #pragma unroll
    for (int j = 0; j < RNK; ++j)
      #pragma unroll
      for (int k = j + 1; k < RNK; ++k) {
        float d = r[j] - r[k];
        // -log sigmoid(d) = max(-d,0) + log1p(exp(-|d|))  (stable softplus)
        acc += fmaxf(-d, 0.0f) + log1pf(expf(-fabsf(d)));
      }
  }
  s[t] = acc;
  __syncthreads();
  for (int off = 512; off >= 1; off >>= 1) {
    if (t < off && t + off < TOK) s[t] += s[t + off];
    __syncthreads();
  }
  if (t == 0) {
    const float npairs = (float)(RNK * (RNK - 1) / 2);  // 36
    out[RNK * BSZ] = s[0] / ((float)TOK * npairs * (float)BSZ);
  }
}

extern "C" void kernel_launch(void* const* d_in, const int* in_sizes, int n_in,
                              void* d_out, int out_size, void* d_ws, size_t ws_size,
                              hipStream_t stream) {
  const float* h  = (const float*)d_in[0];   // [288,576,1024] f32
  const float* w  = (const float*)d_in[1];   // [1,1024] f32
  const float* vb = (const float*)d_in[2];   // [1] f32
  float* out = (float*)d_out;                // 72 end_scores + 1 loss

  float* rewards = (float*)d_ws;                       // 288*576 floats
  float* ranked  = rewards + (size_t)NROW * TOK;       // 72*576 floats

  k_rewards<<<NWAVE / WPB, 256, 0, stream>>>(h, w, vb, rewards);
  k_ranked <<<BSZ * RNK, TOK, 0, stream>>>(rewards, ranked, out);
  k_loss   <<<1, TOK, 0, stream>>>(ranked, out);
}